// VoxelizationByGridShapeDet_47751446397373
// MI455X (gfx1250) — compile-verified
//
#include <hip/hip_runtime.h>
#include <hip/hip_bf16.h>
#include <stdint.h>

// ---------------- problem constants (match reference) ----------------
#define NPTS_MAX   400000
#define CFEAT      4
#define MAXP       32
#define MAXV       150000
#define DX         1024
#define DY         1024
#define DZ         40
#define HASH_BITS  20
#define HASH_SZ    (1u << HASH_BITS)
#define HMASK      (HASH_SZ - 1u)
#define BLK        256

// ---------------- kernel 0: zero/initialize everything (B128 stores) ---------
// out: outN floats (zero). hkey=-1, firstpt=INT_MAX, rankslot=-1 (hashN each).
// vcount=0 (maxv). All counts here are divisible by 4 and 16B-aligned; a
// guarded scalar tail covers any remainder.
__global__ void vox_init_k(float* __restrict__ out, int outN,
                           int* __restrict__ hkey, int* __restrict__ firstpt,
                           int* __restrict__ rankslot, int hashN,
                           int* __restrict__ vcount, int maxv) {
  int tid = blockIdx.x * blockDim.x + threadIdx.x;
  int stride = gridDim.x * blockDim.x;

  const int4 z4  = make_int4(0, 0, 0, 0);
  const int4 m14 = make_int4(-1, -1, -1, -1);
  const int4 mx4 = make_int4(0x7fffffff, 0x7fffffff, 0x7fffffff, 0x7fffffff);

  int outQ = outN >> 2;
  int4* __restrict__ out4 = (int4*)out;
  for (int j = tid; j < outQ; j += stride) out4[j] = z4;          // global_store_b128
  for (int j = (outQ << 2) + tid; j < outN; j += stride) out[j] = 0.0f;

  int hashQ = hashN >> 2;
  int4* __restrict__ hk4 = (int4*)hkey;
  int4* __restrict__ fp4 = (int4*)firstpt;
  int4* __restrict__ rs4 = (int4*)rankslot;
  for (int j = tid; j < hashQ; j += stride) {
    hk4[j] = m14;
    fp4[j] = mx4;
    rs4[j] = m14;
  }
  for (int j = (hashQ << 2) + tid; j < hashN; j += stride) {
    hkey[j] = -1; firstpt[j] = 0x7fffffff; rankslot[j] = -1;
  }

  int vQ = maxv >> 2;
  int4* __restrict__ vc4 = (int4*)vcount;
  for (int j = tid; j < vQ; j += stride) vc4[j] = z4;
  for (int j = (vQ << 2) + tid; j < maxv; j += stride) vcount[j] = 0;
}

// ---------------- kernel 1: voxel id + hash insert + first-point atomicMin ----
__global__ void vox_build_k(const float4* __restrict__ pts,
                            int* __restrict__ lin, int* __restrict__ slot,
                            int* __restrict__ hkey, int* __restrict__ firstpt,
                            int n) {
  int i = blockIdx.x * BLK + threadIdx.x;
  if (i >= n) return;
  // gfx1250 prefetch path: global_prefetch_b8 for the stream ahead of us
  int pf = (i + 4096 < n) ? i + 4096 : i;
  __builtin_prefetch(&pts[pf], 0, 1);

  float4 p = pts[i];
  int cx = (int)floorf((p.x + 51.2f) * 10.0f);
  int cy = (int)floorf((p.y + 51.2f) * 10.0f);
  int cz = (int)floorf((p.z +  5.0f) *  5.0f);
  bool ok = (cx >= 0) & (cx < DX) & (cy >= 0) & (cy < DY) & (cz >= 0) & (cz < DZ);
  if (!ok) { lin[i] = -1; slot[i] = -1; return; }
  int l = (cz << 20) | (cy << 10) | cx;   // (cz*DY + cy)*DX + cx, DX=DY=1024
  lin[i] = l;

  unsigned h = ((unsigned)l * 2654435761u) & HMASK;
  for (;;) {
    int prev = atomicCAS(&hkey[h], -1, l);
    if (prev == -1 || prev == l) break;
    h = (h + 1u) & HMASK;
  }
  slot[i] = (int)h;
  atomicMin(&firstpt[h], i);              // order-independent => deterministic
}

// ---------------- kernel 2: per-block head-flag reduction (wave32 ballot) ----
__global__ void vox_heads_k(const int* __restrict__ lin, const int* __restrict__ slot,
                            const int* __restrict__ firstpt,
                            int* __restrict__ bsums, int n) {
  int t = threadIdx.x;
  int i = blockIdx.x * BLK + t;
  int head = 0;
  if (i < n) {
    int l = lin[i];
    if (l >= 0 && firstpt[slot[i]] == i) head = 1;
  }
  unsigned long long m = __ballot(head);   // wave32: low 32 bits populated
  __shared__ int wsum[BLK / 32];
  if ((t & 31) == 0) wsum[t >> 5] = __popcll(m);
  __syncthreads();
  if (t == 0) {
    int s = 0;
#pragma unroll
    for (int w = 0; w < BLK / 32; ++w) s += wsum[w];
    bsums[blockIdx.x] = s;
  }
}

// ---------------- kernel 3: serial exclusive scan of block sums (1563 ints) --
__global__ void vox_scan_k(int* __restrict__ bsums, int nb) {
  if (blockIdx.x == 0 && threadIdx.x == 0) {
    int acc = 0;
    for (int b = 0; b < nb; ++b) { int v = bsums[b]; bsums[b] = acc; acc += v; }
  }
}

// ---------------- kernel 4: assign first-appearance ranks + write coors ------
__global__ void vox_assign_k(const int* __restrict__ lin, const int* __restrict__ slot,
                             const int* __restrict__ firstpt, const int* __restrict__ bsums,
                             int* __restrict__ rankslot, int* __restrict__ coors, int n) {
  __shared__ int sh[BLK];
  int t = threadIdx.x;
  int i = blockIdx.x * BLK + t;
  int head = 0, s = -1, l = -1;
  if (i < n) {
    l = lin[i];
    if (l >= 0) { s = slot[i]; if (firstpt[s] == i) head = 1; }
  }
  sh[t] = head;
  __syncthreads();
  for (int off = 1; off < BLK; off <<= 1) {     // Hillis-Steele inclusive scan
    int v = (t >= off) ? sh[t - off] : 0;
    __syncthreads();
    sh[t] += v;
    __syncthreads();
  }
  if (head) {
    int rank = bsums[blockIdx.x] + sh[t] - 1;   // exact first-appearance rank
    rankslot[s] = rank;
    if (rank < MAXV) {
      coors[rank * 3 + 0] = l >> 20;            // z
      coors[rank * 3 + 1] = (l >> 10) & 1023;   // y
      coors[rank * 3 + 2] = l & 1023;           // x
    }
  }
}

// ---------------- kernel 5: scatter point rows via async global->LDS stage ---
__global__ void vox_scatter_k(const float4* __restrict__ pts,
                              const int* __restrict__ lin, const int* __restrict__ slot,
                              const int* __restrict__ rankslot,
                              int* __restrict__ vcount, float4* __restrict__ voxels,
                              int n) {
  __shared__ float4 sp[BLK];
  int t = threadIdx.x;
  int i = blockIdx.x * BLK + t;
  int isrc = (i < n) ? i : (n - 1);

  // CDNA5 async data mover path: per-lane B128 global->LDS, tracked by ASYNCcnt
  unsigned ldsa = (unsigned)(unsigned long long)(const void*)&sp[t]; // low 32b = LDS byte addr
  unsigned voff = (unsigned)isrc * 16u;                              // byte offset into pts
  asm volatile("global_load_async_to_lds_b128 %0, %1, %2"
               :: "v"(ldsa), "v"(voff), "s"(pts)
               : "memory");
  asm volatile("s_wait_asynccnt 0x0" ::: "memory");

  if (i < n) {
    int l = lin[i];
    if (l >= 0) {
      int s = slot[i];
      int r = rankslot[s];
      if (r >= 0 && r < MAXV) {
        int p = atomicAdd(&vcount[r], 1);
        if (p < MAXP) voxels[(size_t)r * MAXP + p] = sp[t];  // ds_load_b128 -> global_store_b128
      }
    }
  }
}

// ---------------- kernel 6: clamp per-voxel counts ---------------------------
__global__ void vox_counts_k(const int* __restrict__ vcount, int* __restrict__ nump) {
  int r = blockIdx.x * BLK + threadIdx.x;
  if (r < MAXV) {
    int c = vcount[r];
    nump[r] = (c < MAXP) ? c : MAXP;
  }
}

// ---------------- host-side launch -------------------------------------------
static inline size_t align16(size_t x) { return (x + 15) & ~(size_t)15; }

extern "C" void kernel_launch(void* const* d_in, const int* in_sizes, int n_in,
                              void* d_out, int out_size, void* d_ws, size_t ws_size,
                              hipStream_t stream) {
  const float* pts_f = (const float*)d_in[0];
  int n = in_sizes[0] / CFEAT;                 // 400000

  // output layout: voxels[MAXV][32][4] f32, then coors[MAXV][3] i32, then num[MAXV] i32
  float* out    = (float*)d_out;
  float* voxels = out;
  int*   coors  = (int*)(out + (size_t)MAXV * MAXP * CFEAT);
  int*   nump   = coors + (size_t)MAXV * 3;

  // workspace carve-up (~16.4 MB)
  char* w = (char*)d_ws;
  int* lin      = (int*)w;  w += align16((size_t)NPTS_MAX * 4);
  int* slot     = (int*)w;  w += align16((size_t)NPTS_MAX * 4);
  int* hkey     = (int*)w;  w += align16((size_t)HASH_SZ * 4);
  int* firstpt  = (int*)w;  w += align16((size_t)HASH_SZ * 4);
  int* rankslot = (int*)w;  w += align16((size_t)HASH_SZ * 4);
  int* vcount   = (int*)w;  w += align16((size_t)MAXV * 4);
  int* bsums    = (int*)w;  w += align16(((size_t)(NPTS_MAX + BLK - 1) / BLK) * 4);
  (void)ws_size; (void)n_in;

  int nb = (n + BLK - 1) / BLK;

  vox_init_k<<<2048, BLK, 0, stream>>>(out, out_size,
                                       hkey, firstpt, rankslot, (int)HASH_SZ,
                                       vcount, MAXV);
  vox_build_k<<<nb, BLK, 0, stream>>>((const float4*)pts_f, lin, slot, hkey, firstpt, n);
  vox_heads_k<<<nb, BLK, 0, stream>>>(lin, slot, firstpt, bsums, n);
  vox_scan_k<<<1, 32, 0, stream>>>(bsums, nb);
  vox_assign_k<<<nb, BLK, 0, stream>>>(lin, slot, firstpt, bsums, rankslot, coors, n);
  vox_scatter_k<<<nb, BLK, 0, stream>>>((const float4*)pts_f, lin, slot, rankslot,
                                        vcount, (float4*)voxels, n);
  vox_counts_k<<<(MAXV + BLK - 1) / BLK, BLK, 0, stream>>>(vcount, nump);
}